// seq2seq_31396210934163
// MI455X (gfx1250) — compile-verified
//
#include <hip/hip_runtime.h>
#include <hip/hip_bf16.h>

// ---------------- problem constants ----------------
#define T_    200
#define N_    128
#define NJ_   22
#define H_    512
#define HD_   512
#define E_    512
#define V_    10000
#define K_    6
#define HI_   1536      // E + HD + H
#define TRG_  24
#define TN_   (T_*N_)   // 25600

typedef __attribute__((ext_vector_type(2))) float v2f;
typedef __attribute__((ext_vector_type(8))) float v8f;

__device__ const int PART_IDX[K_][6] = {
    {0, 0, 0, 0, 0, 0},
    {0, 3, 6, 9, 12, 15},
    {14, 17, 19, 21, 0, 0},
    {13, 16, 18, 20, 0, 0},
    {2, 5, 8, 11, 0, 0},
    {1, 4, 7, 10, 0, 0}};
__device__ const int PART_LEN[K_] = {1, 6, 4, 4, 4, 4};

__device__ __forceinline__ float sigm(float x) { return 1.0f / (1.0f + __expf(-x)); }

#define WMMA_F32(a, b, c) \
    __builtin_amdgcn_wmma_f32_16x16x4_f32(false, (a), false, (b), (short)0, (c), false, false)

// ---------------- generic f32 WMMA GEMM ----------------
// out[off + r*ldo_row + c*ldo_col] = act( A[r,:K]·W[c,:K] + bias[c] (+ addsrc[same]) )
// One wave computes a 16x64 C tile: 4 column-tiles = 4 independent WMMA chains,
// 4x reuse of each A fragment. Inner loop is branch-free (OOB rows clamped to row 0,
// polluted C rows/cols masked at writeback) and software-pipelined one K-step deep
// so the next chunk's b64 loads are in flight while the current WMMAs execute.
__global__ void wmma_gemm_f32_kernel(
    const float* __restrict__ A, int lda,
    const float* __restrict__ W, int ldw,
    const float* __restrict__ bias,
    const float* __restrict__ addsrc,
    float* __restrict__ out, long ldo_row, int ldo_col, long out_off,
    int M, int Nd, int Kd, int act)
{
    const int m0   = blockIdx.x * 16;
    const int n0   = blockIdx.y * 64;
    const int lane = threadIdx.x & 31;
    const int half = lane >> 4;      // 0: K=0,1  1: K=2,3  (ISA 7.12.2 striping)
    const int sub  = lane & 15;

    const int ar  = m0 + sub;        // A row owned by this lane
    const bool av = ar < M;
    const float* Ar = A + (long)(av ? ar : 0) * lda;

    const float* Wr[4];
    bool wv[4];
#pragma unroll
    for (int j = 0; j < 4; ++j) {
        const int wr = n0 + 16 * j + sub;   // W row (output column) for col-tile j
        wv[j] = wr < Nd;
        Wr[j] = W + (long)(wv[j] ? wr : 0) * ldw;
    }

    v8f acc[4];
#pragma unroll
    for (int j = 0; j < 4; ++j) acc[j] = (v8f){0.f,0.f,0.f,0.f,0.f,0.f,0.f,0.f};

    const int kb = 2 * half;
    if ((Kd & 3) == 0 && Kd >= 8) {
        // ---- software-pipelined path (all shapes in this model) ----
        v2f aC  = *(const v2f*)(Ar + kb);
        v2f bC0 = *(const v2f*)(Wr[0] + kb);
        v2f bC1 = *(const v2f*)(Wr[1] + kb);
        v2f bC2 = *(const v2f*)(Wr[2] + kb);
        v2f bC3 = *(const v2f*)(Wr[3] + kb);
        for (int k = 0; k + 8 <= Kd; k += 4) {
            const int kn = k + 4 + kb;
            const v2f aN  = *(const v2f*)(Ar + kn);       // next chunk in flight
            const v2f bN0 = *(const v2f*)(Wr[0] + kn);
            const v2f bN1 = *(const v2f*)(Wr[1] + kn);
            const v2f bN2 = *(const v2f*)(Wr[2] + kn);
            const v2f bN3 = *(const v2f*)(Wr[3] + kn);
            acc[0] = WMMA_F32(aC, bC0, acc[0]);           // consume current chunk
            acc[1] = WMMA_F32(aC, bC1, acc[1]);
            acc[2] = WMMA_F32(aC, bC2, acc[2]);
            acc[3] = WMMA_F32(aC, bC3, acc[3]);
            aC = aN; bC0 = bN0; bC1 = bN1; bC2 = bN2; bC3 = bN3;
        }
        acc[0] = WMMA_F32(aC, bC0, acc[0]);               // epilogue chunk
        acc[1] = WMMA_F32(aC, bC1, acc[1]);
        acc[2] = WMMA_F32(aC, bC2, acc[2]);
        acc[3] = WMMA_F32(aC, bC3, acc[3]);
    } else {
        // ---- generic guarded path (any K) ----
        int k = 0;
        for (; k + 4 <= Kd; k += 4) {
            const int ka = k + kb;
            const v2f a = *(const v2f*)(Ar + ka);
#pragma unroll
            for (int j = 0; j < 4; ++j) {
                const v2f b = *(const v2f*)(Wr[j] + ka);
                acc[j] = WMMA_F32(a, b, acc[j]);
            }
        }
        if (k < Kd) {
            const int ka = k + kb;
            v2f a = {0.f, 0.f};
            if (ka     < Kd) a.x = Ar[ka];
            if (ka + 1 < Kd) a.y = Ar[ka + 1];
#pragma unroll
            for (int j = 0; j < 4; ++j) {
                v2f b = {0.f, 0.f};
                if (ka     < Kd) b.x = Wr[j][ka];
                if (ka + 1 < Kd) b.y = Wr[j][ka + 1];
                acc[j] = WMMA_F32(a, b, acc[j]);
            }
        }
    }

#pragma unroll
    for (int j = 0; j < 4; ++j) {
        const int c = n0 + 16 * j + sub;
        if (!wv[j]) continue;
#pragma unroll
        for (int v = 0; v < 8; ++v) {      // C/D layout: vgpr v -> row v (+8 upper half)
            const int r = m0 + v + 8 * half;
            if (r < M) {
                float val = acc[j][v];
                if (bias) val += bias[c];
                const long o = out_off + (long)r * ldo_row + (long)c * ldo_col;
                if (addsrc) val += addsrc[o];
                if (act == 1) val = tanhf(val);
                out[o] = val;
            }
        }
    }
}

// ---------------- encoder layer 1 (tiny inner dim, plain VALU) ----------------
__global__ void enc_layer1_kernel(const float* __restrict__ x,
                                  const float* __restrict__ W1,
                                  const float* __restrict__ b1,
                                  float* __restrict__ h1,
                                  int part, int vstream)
{
    long i = (long)blockIdx.x * blockDim.x + threadIdx.x;
    if (i >= (long)TN_ * HD_) return;
    const int  c  = (int)(i % HD_);
    const long tn = i / HD_;
    const int  t  = (int)(tn / N_);
    const int  len = PART_LEN[part];
    const int  d   = len * 3;
    const float* xr  = x + tn * (NJ_ * 3);
    const float* xrp = x + (tn - N_) * (NJ_ * 3);
    float sum = b1[c];
    for (int j = 0; j < len; ++j) {
        const int jj = PART_IDX[part][j];
        for (int a = 0; a < 3; ++a) {
            float xv = xr[jj * 3 + a];
            if (vstream) xv = (t > 0) ? (xv - xrp[jj * 3 + a]) : 0.f;
            sum += xv * W1[c * d + j * 3 + a];
        }
    }
    h1[tn * HD_ + c] = tanhf(sum);
}

// ---------------- misc small kernels ----------------
__global__ void zero_kernel(float* __restrict__ p, long n)
{
    long i = (long)blockIdx.x * blockDim.x + threadIdx.x;
    if (i < n) p[i] = 0.f;
}

__global__ void emb_gather_kernel(const float* __restrict__ emb,
                                  const int* __restrict__ y, int jrow,
                                  float* __restrict__ embt)
{
    int i = blockIdx.x * blockDim.x + threadIdx.x;
    if (i >= N_ * E_) return;
    const int n = i / E_, e = i % E_;
    const int tok = y[jrow * N_ + n];
    embt[i] = emb[(long)tok * E_ + e];
}

__global__ void lstm_point_kernel(const float* __restrict__ g,
                                  float* __restrict__ h, float* __restrict__ c)
{
    int i = blockIdx.x * blockDim.x + threadIdx.x;
    if (i >= N_ * H_) return;
    const int n = i / H_, hh = i % H_;
    const float* gr = g + (long)n * 4 * H_;
    const float ig = sigm(gr[hh]);
    const float fg = sigm(gr[H_ + hh]);
    const float gg = tanhf(gr[2 * H_ + hh]);
    const float og = sigm(gr[3 * H_ + hh]);
    const float c2 = fg * c[i] + ig * gg;
    c[i] = c2;
    h[i] = og * tanhf(c2);
}

// spatial scores (6) + temporal score (1) per (t,n); one 64-thread block each
__global__ void attn_scores_kernel(const float* __restrict__ fxR,
                                   const float* __restrict__ fgR,
                                   const float* __restrict__ fh,
                                   const float* __restrict__ tfh,
                                   const float* __restrict__ spatW,
                                   const float* __restrict__ spatB,
                                   const float* __restrict__ tattW,
                                   const float* __restrict__ tattB,
                                   float* __restrict__ sbuf,
                                   float* __restrict__ epbuf)
{
    const int tn  = blockIdx.x;
    const int n   = tn % N_;
    const int tid = threadIdx.x;      // 64 threads = 2 waves
    const float* fx  = fxR + (long)tn * HD_;
    const float* fg  = fgR + (long)tn * HD_;
    const float* fhn = fh  + (long)n  * HD_;
    const float* tfn = tfh + (long)n  * HD_;
    float p[7] = {0.f, 0.f, 0.f, 0.f, 0.f, 0.f, 0.f};
    for (int c = tid; c < HD_; c += 64) {
        const float u = tanhf(fhn[c] + fx[c]);
#pragma unroll
        for (int kk = 0; kk < K_; ++kk) p[kk] += u * spatW[kk * HD_ + c];
        const float v = tanhf(tfn[c] + fg[c]);
        p[6] += v * tattW[c];
    }
    __shared__ float red[7][64];
    for (int kk = 0; kk < 7; ++kk) red[kk][tid] = p[kk];
    __syncthreads();
    for (int st = 32; st > 0; st >>= 1) {
        if (tid < st)
            for (int kk = 0; kk < 7; ++kk) red[kk][tid] += red[kk][tid + st];
        __syncthreads();
    }
    if (tid == 0) {
        for (int kk = 0; kk < K_; ++kk)
            sbuf[(long)tn * K_ + kk] = red[kk][0] + spatB[kk];
        epbuf[tn] = red[6][0] + tattB[0];
    }
}

// softmax over K=6 (in place) * mask
__global__ void alpha_kernel(float* __restrict__ sbuf, const int* __restrict__ src_lens)
{
    int tn = blockIdx.x * blockDim.x + threadIdx.x;
    if (tn >= TN_) return;
    const int n = tn % N_, t = tn / N_;
    float* s = sbuf + (long)tn * K_;
    float m = s[0];
#pragma unroll
    for (int k = 1; k < K_; ++k) m = fmaxf(m, s[k]);
    float e[K_], sum = 0.f;
#pragma unroll
    for (int k = 0; k < K_; ++k) { e[k] = __expf(s[k] - m); sum += e[k]; }
    const float mask = (t < src_lens[n]) ? 1.f : 0.f;
    const float inv  = mask / sum;
#pragma unroll
    for (int k = 0; k < K_; ++k) s[k] = e[k] * inv;
}

// per-column n: masked softmax over T, Gaussian window
__global__ void temporal_kernel(const float* __restrict__ ep,
                                const int* __restrict__ src_lens,
                                float* __restrict__ window)
{
    const int n = blockIdx.x;
    const int t = threadIdx.x;                 // 256 threads, T_=200
    __shared__ float sh[256];
    const int  sl    = src_lens[n];
    const bool valid = (t < T_) && (t < sl);
    const float val  = valid ? ep[(long)t * N_ + n] : -3.0e38f;

    sh[t] = val; __syncthreads();
    for (int st = 128; st > 0; st >>= 1) { if (t < st) sh[t] = fmaxf(sh[t], sh[t + st]); __syncthreads(); }
    const float mx = sh[0]; __syncthreads();

    const float e = valid ? __expf(val - mx) : 0.f;
    sh[t] = e; __syncthreads();
    for (int st = 128; st > 0; st >>= 1) { if (t < st) sh[t] += sh[t + st]; __syncthreads(); }
    const float sum = sh[0]; __syncthreads();
    const float b = e / sum;

    sh[t] = b * (float)t; __syncthreads();
    for (int st = 128; st > 0; st >>= 1) { if (t < st) sh[t] += sh[t + st]; __syncthreads(); }
    const float mean = sh[0]; __syncthreads();

    const float d = (float)t - mean;
    sh[t] = b * d * d; __syncthreads();
    for (int st = 128; st > 0; st >>= 1) { if (t < st) sh[t] += sh[t + st]; __syncthreads(); }
    const float stdv = sqrtf(sh[0]);

    if (t < T_) {
        const float dd = mean - (float)t;
        window[(long)t * N_ + n] = __expf(-(dd * dd) / (2.f * stdv * stdv));
    }
}

// ctx[n,h] = (1/K) * sum_t window[t,n] * sum_k xparts[t,n,h,k]*alpha[t,n,k]
__global__ void ctx_kernel(const float* __restrict__ xparts,
                           const float* __restrict__ alpha,
                           const float* __restrict__ window,
                           float* __restrict__ ctx)
{
    int i = blockIdx.x * blockDim.x + threadIdx.x;
    if (i >= N_ * H_) return;
    const int n = i / H_, h = i % H_;
    float acc = 0.f;
    for (int t = 0; t < T_; ++t) {
        const long tn = (long)t * N_ + n;
        const float w = window[tn];
        const float* al = alpha  + tn * K_;
        const float* xp = xparts + (tn * H_ + h) * K_;
        if (t + 1 < T_)   // exercise global_prefetch_b8 on the streamed tensor
            __builtin_prefetch(xparts + (((long)(t + 1) * N_ + n) * H_ + h) * K_, 0, 1);
        float sv = 0.f;
#pragma unroll
        for (int k = 0; k < K_; ++k) sv += xp[k] * al[k];
        acc += w * sv;
    }
    ctx[i] = acc * (1.f / (float)K_);
}

__global__ void beta_kernel(const float* __restrict__ bh, const float* __restrict__ embt,
                            const float* __restrict__ gW, const float* __restrict__ gB,
                            const float* __restrict__ aW, const float* __restrict__ aB,
                            float* __restrict__ beta)
{
    int n = blockIdx.x * blockDim.x + threadIdx.x;
    if (n >= N_) return;
    float s = gB[0] + aB[0];
    for (int c = 0; c < H_; ++c)
        s += gW[c] * bh[(long)n * H_ + c] + aW[c] * embt[(long)n * E_ + c];
    beta[n] = sigm(s);
}

// acv = beta*ctx_t + (1-beta)*ht_t ; cat = [emb, acv, bh]
__global__ void combine_kernel(const float* __restrict__ beta,
                               const float* __restrict__ ctxt, const float* __restrict__ htt,
                               const float* __restrict__ embt, const float* __restrict__ bh,
                               float* __restrict__ cat)
{
    int i = blockIdx.x * blockDim.x + threadIdx.x;
    if (i >= N_ * H_) return;
    const int n = i / H_, h = i % H_;
    const float bt = beta[n];
    float* cr = cat + (long)n * HI_;
    cr[h]             = embt[i];
    cr[E_ + h]        = bt * ctxt[i] + (1.f - bt) * htt[i];
    cr[E_ + HD_ + h]  = bh[i];
}

// ---------------- host side ----------------
static inline void gemm(hipStream_t s, const float* A, int lda, const float* W, int ldw,
                        const float* bias, const float* addsrc,
                        float* out, long lr, int lc, long off,
                        int M, int Nd, int Kd, int act)
{
    dim3 g((M + 15) / 16, (Nd + 63) / 64);
    wmma_gemm_f32_kernel<<<g, 32, 0, s>>>(A, lda, W, ldw, bias, addsrc,
                                          out, lr, lc, off, M, Nd, Kd, act);
}

extern "C" void kernel_launch(void* const* d_in, const int* in_sizes, int n_in,
                              void* d_out, int out_size, void* d_ws, size_t ws_size,
                              hipStream_t stream)
{
    (void)in_sizes; (void)n_in; (void)out_size; (void)ws_size;

    // setup_inputs() insertion order, depth-first through params
    const float* x        = (const float*)d_in[0];
    const int*   y        = (const int*)d_in[1];
    const int*   src_lens = (const int*)d_in[2];
    auto F = [&](int i) { return (const float*)d_in[i]; };
    // enc: indices 3..50 (6 parts x 8 arrays)
    const float* emb   = F(51);
    const float* bWih = F(52); const float* bWhh = F(53);
    const float* bbih = F(54); const float* bbhh = F(55);
    const float* tWih = F(56); const float* tWhh = F(57);
    const float* tbih = F(58); const float* tbhh = F(59);
    const float* fxW  = F(60); const float* fxb  = F(61);
    const float* fhW  = F(62); const float* fhb  = F(63);
    const float* spW  = F(64); const float* spb  = F(65);
    const float* fgW  = F(66); const float* fgb  = F(67);
    const float* tfW  = F(68); const float* tfb  = F(69);
    const float* taW  = F(70); const float* tab  = F(71);
    const float* gaW  = F(72); const float* gab  = F(73);
    const float* adW  = F(74); const float* adb  = F(75);
    const float* ctW  = F(76); const float* ctb  = F(77);
    const float* htW  = F(78); const float* htb  = F(79);
    const float* mxW  = F(80); const float* mxb  = F(81);
    const float* fiW  = F(82); const float* fib  = F(83);
    float* out = (float*)d_out;

    // workspace carve-up
    float* ws = (float*)d_ws;
    size_t o = 0;
    float* xparts = ws + o; o += (size_t)TN_ * H_ * K_;   // (T,N,H,K) == R (T,N,3072)
    float* h1     = ws + o; o += (size_t)TN_ * HD_;
    float* fxR    = ws + o; o += (size_t)TN_ * HD_;
    float* fgR    = ws + o; o += (size_t)TN_ * HD_;
    float* sbuf   = ws + o; o += (size_t)TN_ * K_;
    float* epbuf  = ws + o; o += (size_t)TN_;
    float* winbuf = ws + o; o += (size_t)TN_;
    float* state  = ws + o;                                // bh,bc,th,tc contiguous
    float* bh = state;            float* bc = state + (size_t)N_ * H_;
    float* th = state + 2ul * N_ * H_; float* tc = state + 3ul * N_ * H_;
    o += 4ul * N_ * H_;
    float* gbuf  = ws + o; o += (size_t)N_ * 4 * H_;
    float* embt  = ws + o; o += (size_t)N_ * E_;
    float* fh    = ws + o; o += (size_t)N_ * HD_;
    float* tfh   = ws + o; o += (size_t)N_ * HD_;
    float* ctx   = ws + o; o += (size_t)N_ * H_;
    float* ctxt  = ws + o; o += (size_t)N_ * HD_;
    float* htt   = ws + o; o += (size_t)N_ * HD_;
    float* betab = ws + o; o += (size_t)N_;
    float* catb  = ws + o; o += (size_t)N_ * HI_;
    float* mixo  = ws + o; o += (size_t)N_ * HI_;

    // ---- encoders: layer1 (VALU) -> layer2 (WMMA, strided into xparts) ----
    const long l1n = (long)TN_ * HD_;
    for (int p = 0; p < K_; ++p) {
        for (int s = 0; s < 2; ++s) {
            const int base = 3 + p * 8 + (s ? 4 : 0);
            enc_layer1_kernel<<<dim3((unsigned)((l1n + 255) / 256)), 256, 0, stream>>>(
                x, F(base), F(base + 1), h1, p, s);
            // h1 (25600x512) @ W2^T (256x512) -> xparts[:, s*256 + c, p]
            gemm(stream, h1, HD_, F(base + 2), HD_, F(base + 3), nullptr,
                 xparts, (long)H_ * K_, K_, (long)(s * 256) * K_ + p,
                 TN_, H_ / 2, HD_, /*tanh*/1);
        }
    }
    // loop-invariant attention projections: R (T*N,3072)
    gemm(stream, xparts, H_ * K_, fxW, H_ * K_, fxb, nullptr, fxR, HD_, 1, 0, TN_, HD_, H_ * K_, 0);
    gemm(stream, xparts, H_ * K_, fgW, H_ * K_, fgb, nullptr, fgR, HD_, 1, 0, TN_, HD_, H_ * K_, 0);

    // zero decoder state (every call: deterministic replay)
    zero_kernel<<<dim3((4u * N_ * H_ + 255) / 256), 256, 0, stream>>>(state, 4l * N_ * H_);

    // ---- decode steps ----
    for (int j = 0; j < TRG_ - 1; ++j) {
        emb_gather_kernel<<<dim3((N_ * E_ + 255) / 256), 256, 0, stream>>>(emb, y, j, embt);

        // bottom LSTM: g = emb@Wih^T + bih + bh@Whh^T + bhh
        gemm(stream, embt, E_, bWih, E_, bbih, nullptr, gbuf, 4 * H_, 1, 0, N_, 4 * H_, E_, 0);
        gemm(stream, bh,   H_, bWhh, H_, bbhh, gbuf,    gbuf, 4 * H_, 1, 0, N_, 4 * H_, H_, 0);
        lstm_point_kernel<<<dim3((N_ * H_ + 255) / 256), 256, 0, stream>>>(gbuf, bh, bc);

        // top LSTM: input = new bh
        gemm(stream, bh, H_, tWih, H_, tbih, nullptr, gbuf, 4 * H_, 1, 0, N_, 4 * H_, H_, 0);
        gemm(stream, th, H_, tWhh, H_, tbhh, gbuf,    gbuf, 4 * H_, 1, 0, N_, 4 * H_, H_, 0);
        lstm_point_kernel<<<dim3((N_ * H_ + 255) / 256), 256, 0, stream>>>(gbuf, th, tc);

        // decoder-state projections
        gemm(stream, bh, H_, fhW, H_, fhb, nullptr, fh,  HD_, 1, 0, N_, HD_, H_, 0);
        gemm(stream, bh, H_, tfW, H_, tfb, nullptr, tfh, HD_, 1, 0, N_, HD_, H_, 0);

        // attention scores (fused spatial + temporal tanh passes)
        attn_scores_kernel<<<dim3(TN_), 64, 0, stream>>>(fxR, fgR, fh, tfh,
                                                         spW, spb, taW, tab, sbuf, epbuf);
        alpha_kernel<<<dim3((TN_ + 255) / 256), 256, 0, stream>>>(sbuf, src_lens);
        temporal_kernel<<<dim3(N_), 256, 0, stream>>>(epbuf, src_lens, winbuf);
        ctx_kernel<<<dim3((N_ * H_ + 255) / 256), 256, 0, stream>>>(xparts, sbuf, winbuf, ctx);

        beta_kernel<<<dim3(1), 128, 0, stream>>>(bh, embt, gaW, gab, adW, adb, betab);
        gemm(stream, ctx, H_, ctW, H_, ctb, nullptr, ctxt, HD_, 1, 0, N_, HD_, H_, 1);
        gemm(stream, th,  H_, htW, H_, htb, nullptr, htt,  HD_, 1, 0, N_, HD_, H_, 1);
        combine_kernel<<<dim3((N_ * H_ + 255) / 256), 256, 0, stream>>>(betab, ctxt, htt,
                                                                        embt, bh, catb);

        // mix (tanh) then fin -> logits directly into d_out
        gemm(stream, catb, HI_, mxW, HI_, mxb, nullptr, mixo, HI_, 1, 0, N_, HI_, HI_, 1);
        gemm(stream, mixo, HI_, fiW, HI_, fib, nullptr, out, V_, 1,
             (long)j * N_ * V_, N_, V_, HI_, 0);
    }
}